// SparseCoCN_76459007803895
// MI455X (gfx1250) — compile-verified
//
#include <hip/hip_runtime.h>

// Problem constants from the reference:
// B=4, H=4, N=1024, F=4, E=131072; out = (B*H, N, N, F) f32 = 256 MB.
#define B_ 4
#define H_ 4
#define N_ 1024
#define F_ 4
#define E_ 131072

typedef __attribute__((ext_vector_type(4))) float v4f;
typedef __attribute__((ext_vector_type(4))) int v4i;

// ---------------------------------------------------------------------------
// Scatter kernel (first in file so the asm snippet shows it): one thread per
// edge; unrolled loop over H heads. Edge streams are read-once -> NT load
// hints; the 2x64KB perm tables are re-read by every block -> regular loads
// (stay hot in WGP$/L2). 4 native f32 global atomics per (edge, head).
// ---------------------------------------------------------------------------
__global__ __launch_bounds__(256) void scatter_kernel(
    const float* __restrict__ perm_val,   // (B, H, N) f32
    const int* __restrict__ perm_tgt,     // (B, H, N) i32
    const int* __restrict__ edge_batch,   // (E,) i32
    const int* __restrict__ edge_src,     // (E,) i32
    const int* __restrict__ edge_tgt,     // (E,) i32
    const float* __restrict__ edge_vals,  // (E, F) f32
    float* __restrict__ out)              // (B*H, N, N, F) f32
{
  const int e = blockIdx.x * blockDim.x + threadIdx.x;
  if (e >= E_) return;

  // Read-once edge streams: non-temporal loads (th:TH_LOAD_NT).
  const int b = __builtin_nontemporal_load(edge_batch + e);
  const int s = __builtin_nontemporal_load(edge_src + e);
  const int t = __builtin_nontemporal_load(edge_tgt + e);
  const v4f ev =
      __builtin_nontemporal_load((const v4f*)(edge_vals + (size_t)e * F_));

#pragma unroll
  for (int h = 0; h < H_; ++h) {
    const size_t tbl = ((size_t)b * H_ + h) * N_;
    const float* pv = perm_val + tbl;
    const int*   pt = perm_tgt + tbl;

    const float sc = pv[s] * pv[t];          // mul_l * mul_r
    const int ns = pt[s];                    // node_src
    const int nt = pt[t];                    // node_tgt

    // batch_idx = b + B*h ; out[batch_idx, ns, nt, 0..3] += ev * sc
    const size_t base =
        ((((size_t)(b + B_ * h)) * N_ + (size_t)ns) * N_ + (size_t)nt) * F_;
    float* o = out + base;
    unsafeAtomicAdd(o + 0, ev.x * sc);
    unsafeAtomicAdd(o + 1, ev.y * sc);
    unsafeAtomicAdd(o + 2, ev.z * sc);
    unsafeAtomicAdd(o + 3, ev.w * sc);
  }
}

// ---------------------------------------------------------------------------
// Zero the 256 MB output with non-temporal 128-bit stores, 4x unrolled
// grid-stride (each unrolled store instruction stays lane-contiguous).
// NT keeps the zero-fill from evicting the atomic working set in L2.
// ---------------------------------------------------------------------------
__global__ __launch_bounds__(256) void zero_out_kernel(float* __restrict__ out,
                                                       long long n4) {
  long long i = (long long)blockIdx.x * blockDim.x + threadIdx.x;
  const long long stride = (long long)gridDim.x * blockDim.x;
  v4f* o = (v4f*)out;
  v4f z = (v4f)0.0f;
  for (; i + 3 * stride < n4; i += 4 * stride) {
    __builtin_nontemporal_store(z, o + i);
    __builtin_nontemporal_store(z, o + i + stride);
    __builtin_nontemporal_store(z, o + i + 2 * stride);
    __builtin_nontemporal_store(z, o + i + 3 * stride);
  }
  for (; i < n4; i += stride) {
    __builtin_nontemporal_store(z, o + i);
  }
}

// ---------------------------------------------------------------------------
// Availability probe (never launched): confirms the gfx1250 async global->LDS
// path lowers on this toolchain (global_load_async_to_lds_b128 +
// s_wait_asynccnt). Not used on the hot path: the edge streams are already
// perfectly coalesced 16B/lane loads, so a global->LDS->VGPR round-trip would
// only add latency for this scatter-bound workload.
// ---------------------------------------------------------------------------
#if defined(__HIP_DEVICE_COMPILE__) &&                                   \
    __has_builtin(__builtin_amdgcn_global_load_async_to_lds_b128) &&     \
    __has_builtin(__builtin_amdgcn_s_wait_asynccnt)
typedef __attribute__((address_space(1))) v4i* gv4i_p;
typedef __attribute__((address_space(3))) v4i* sv4i_p;

__global__ void probe_async_lds_kernel(const float* __restrict__ g,
                                       float* __restrict__ o) {
  __shared__ float sh[1024];
  const unsigned tid = threadIdx.x;
  __builtin_amdgcn_global_load_async_to_lds_b128(
      (gv4i_p)(const void*)(g + tid * 4),
      (sv4i_p)(void*)(sh + tid * 4),
      /*offset=*/0, /*cpol=*/0);
  __builtin_amdgcn_s_wait_asynccnt(0);
  __syncthreads();
  o[tid] = sh[tid];
}
#endif

// ---------------------------------------------------------------------------
// Host launcher
// ---------------------------------------------------------------------------
extern "C" void kernel_launch(void* const* d_in, const int* in_sizes, int n_in,
                              void* d_out, int out_size, void* d_ws,
                              size_t ws_size, hipStream_t stream) {
  (void)n_in; (void)d_ws; (void)ws_size;

  const float* perm_val   = (const float*)d_in[0];
  const int*   perm_tgt   = (const int*)d_in[1];
  const int*   edge_batch = (const int*)d_in[2];
  const int*   edge_src   = (const int*)d_in[3];
  const int*   edge_tgt   = (const int*)d_in[4];
  const float* edge_vals  = (const float*)d_in[5];
  float* out = (float*)d_out;

  // 1) zero the output (out_size = B*H*N*N*F = 67,108,864 floats)
  const long long n4 = (long long)out_size / 4;
  zero_out_kernel<<<4096, 256, 0, stream>>>(out, n4);

  // 2) scatter-add: one thread per edge
  const int e = in_sizes[2];  // E
  scatter_kernel<<<(e + 255) / 256, 256, 0, stream>>>(
      perm_val, perm_tgt, edge_batch, edge_src, edge_tgt, edge_vals, out);
}